// Distance_14697378087511
// MI455X (gfx1250) — compile-verified
//
#include <hip/hip_runtime.h>
#include <hip/hip_bf16.h>

#define KNBR 32
#define STOP2 25.0f
#define MAXC 128   // per-node candidate buffer; expected in-radius count ~5

typedef float v2f __attribute__((ext_vector_type(2)));
typedef float v8f __attribute__((ext_vector_type(8)));

// One wave (32 threads) per 16-node tile. batch is sorted, so the tile's
// candidate set is the contiguous range [loT, hiT) found by binary search.
// Each 16-candidate sub-tile's 16x16 dot-product block is one
// V_WMMA_F32_16X16X4_F32 (K = x,y,z,0). Candidate positions are prefetched
// two tiles ahead with global_prefetch_b8 (no LOADcnt impact), so the b96
// load in the loop hits near cache.
__global__ __launch_bounds__(32) void radius_graph_topk_kernel(
    const float* __restrict__ pos,        // [N,3] f32
    const long long* __restrict__ batch,  // [N] i64, sorted
    float* __restrict__ out,              // 7*N*K floats
    int N)
{
    const int i0   = blockIdx.x * 16;
    const int lane = threadIdx.x;       // 0..31
    const int col  = lane & 15;         // column within 16x16 block
    const int half = lane >> 4;         // 0: rows 0..7 / K=0,1 ; 1: rows 8..15 / K=2,3

    __shared__ float posI[16][3];
    __shared__ float sqI[16];
    __shared__ int   batI[16];
    __shared__ unsigned int cnt[16];
    __shared__ float d2l[16][MAXC];
    __shared__ int   jl[16][MAXC];

    if (lane < 16) {
        int i = i0 + lane;
        int ic = i < N ? i : N - 1;
        float x = pos[3*ic+0], y = pos[3*ic+1], z = pos[3*ic+2];
        posI[lane][0] = x; posI[lane][1] = y; posI[lane][2] = z;
        sqI[lane] = x*x + y*y + z*z;
        batI[lane] = (int)batch[ic];
        cnt[lane] = 0u;
    }
    __syncthreads();

    // A operand (16x4 f32): lanes 0-15 hold {x,y} (K=0,1), lanes 16-31 hold {z,0} (K=2,3)
    v2f A;
    if (half == 0) { A[0] = posI[col][0]; A[1] = posI[col][1]; }
    else           { A[0] = posI[col][2]; A[1] = 0.0f; }

    // Register copies of this lane's 8 D-rows' constants. batR = -1 sentinel
    // for rows past N (batch values are >= 0), so the push condition needs no
    // separate i<N test.
    float sqR[8]; int batR[8];
    for (int r = 0; r < 8; ++r) {
        int m = r + 8*half;
        sqR[r]  = sqI[m];
        batR[r] = (i0 + m < N) ? batI[m] : -1;
    }

    // Wave-uniform candidate range for this tile (batch sorted ascending).
    long long bLo = (long long)batI[0];
    long long bHi = (long long)batI[15];
    int lo = 0, hi = N;
    while (lo < hi) { int mid = (lo + hi) >> 1; if (batch[mid] <  bLo) lo = mid + 1; else hi = mid; }
    const int loT = lo;
    lo = 0; hi = N;
    while (lo < hi) { int mid = (lo + hi) >> 1; if (batch[mid] <= bHi) lo = mid + 1; else hi = mid; }
    const int hiT = lo;

    // Warm the cache for the first two tiles.
    if (loT < hiT) {
        int jp0 = loT + col;          if (jp0 >= hiT) jp0 = hiT - 1;
        int jp1 = loT + 16 + col;     if (jp1 >= hiT) jp1 = hiT - 1;
        __builtin_prefetch(pos + 3*jp0, 0, 3);   // global_prefetch_b8
        __builtin_prefetch(pos + 3*jp1, 0, 3);
    }

    for (int j0 = loT; j0 < hiT; j0 += 16) {
        // Prefetch two tiles ahead (clamped; speculative prefetch is safe).
        {
            int jp = j0 + 32 + col;
            if (jp >= hiT) jp = hiT - 1;
            __builtin_prefetch(pos + 3*jp, 0, 3);
        }

        int  j    = j0 + col;
        bool colv = (j < hiT);
        int  jc   = colv ? j : (hiT - 1);
        float jx = pos[3*jc+0], jy = pos[3*jc+1], jz = pos[3*jc+2];
        float sqJ = jx*jx + jy*jy + jz*jz;

        // B operand (4x16 f32), same per-lane construction as A (Gram is A*A^T).
        v2f B;
        if (half == 0) { B[0] = jx; B[1] = jy; }
        else           { B[0] = jz; B[1] = 0.0f; }

        v8f C = {};   // D[m][n] = dot(pos_i[m], pos_j[n]); EXEC all ones here.
#if defined(__has_builtin) && __has_builtin(__builtin_amdgcn_wmma_f32_16x16x4_f32)
        C = __builtin_amdgcn_wmma_f32_16x16x4_f32(false, A, false, B,
                                                  (short)0, C, false, false);
#else
        for (int r = 0; r < 8; ++r) {
            int m = r + 8*half;
            C[r] = posI[m][0]*jx + posI[m][1]*jy + posI[m][2]*jz;
        }
#endif
        // Early-out: minimum d2 over this lane's 8 rows. If it misses the
        // cutoff (the common case), skip all push machinery for this lane.
        float e0 = fminf(sqR[0] - 2.0f*C[0], sqR[1] - 2.0f*C[1]);
        float e1 = fminf(sqR[2] - 2.0f*C[2], sqR[3] - 2.0f*C[3]);
        float e2 = fminf(sqR[4] - 2.0f*C[4], sqR[5] - 2.0f*C[5]);
        float e3 = fminf(sqR[6] - 2.0f*C[6], sqR[7] - 2.0f*C[7]);
        float mn = fminf(fminf(e0, e1), fminf(e2, e3));

        if (colv && (sqJ + mn) <= STOP2) {
            int bj = (int)batch[j];                 // lazy: only on near columns
            for (int r = 0; r < 8; ++r) {
                int m = r + 8*half;                 // D-layout: VGPR r -> row r (+8 upper half)
                int i = i0 + m;
                float d2 = sqJ + sqR[r] - 2.0f*C[r];
                if (bj == batR[r] && j != i && d2 <= STOP2) {
                    unsigned int s = atomicAdd(&cnt[m], 1u);   // ds_add_rtn_u32
                    if (s < MAXC) { d2l[m][s] = d2; jl[m][s] = j; }
                }
            }
        }
    }
    __syncthreads();

    // Lanes 0-15: exact top-K per node, ordered by (d2, index) ascending.
    if (lane < 16) {
        int m = lane;
        int c = (int)cnt[m]; if (c > MAXC) c = MAXC;
        int kc = c < KNBR ? c : KNBR;
        for (int k = 0; k < kc; ++k) {              // partial selection sort in LDS
            int best = k; float bd = d2l[m][k]; int bj = jl[m][k];
            for (int t = k + 1; t < c; ++t) {
                float d = d2l[m][t]; int jj = jl[m][t];
                if (d < bd || (d == bd && jj < bj)) { best = t; bd = d; bj = jj; }
            }
            if (best != k) {
                d2l[m][best] = d2l[m][k]; jl[m][best] = jl[m][k];
                d2l[m][k] = bd;           jl[m][k] = bj;
            }
        }
    }
    __syncthreads();

    // All 32 lanes emit: lane = (node col, k-half); 16 edges per lane.
    const long long NK = (long long)N * KNBR;
    {
        int m = col;
        int i = i0 + m;
        if (i < N) {
            int c = (int)cnt[m]; if (c > MAXC) c = MAXC;
            int kc = c < KNBR ? c : KNBR;
            float ix = posI[m][0], iy = posI[m][1], iz = posI[m][2];
            for (int kk = 0; kk < 16; ++kk) {
                int k = half * 16 + kk;
                long long e = (long long)i * KNBR + k;
                int nbr = i; float w = 0.f, vx = 0.f, vy = 0.f, vz = 0.f, msk = 0.f;
                if (k < kc) {
                    nbr = jl[m][k];
                    vx = ix - pos[3*nbr+0];
                    vy = iy - pos[3*nbr+1];
                    vz = iz - pos[3*nbr+2];
                    w  = sqrtf(vx*vx + vy*vy + vz*vz);
                    msk = 1.0f;                       // START==0 -> lower mask == validity
                }
                out[e]              = (float)i;       // edge_index[0]
                out[NK + e]         = (float)nbr;     // edge_index[1]
                out[2*NK + e]       = w;              // edge_weight
                out[3*NK + e*3 + 0] = vx;             // edge_vec
                out[3*NK + e*3 + 1] = vy;
                out[3*NK + e*3 + 2] = vz;
                out[6*NK + e]       = msk;            // mask
            }
        }
    }
}

extern "C" void kernel_launch(void* const* d_in, const int* in_sizes, int n_in,
                              void* d_out, int out_size, void* d_ws, size_t ws_size,
                              hipStream_t stream) {
    (void)n_in; (void)out_size; (void)d_ws; (void)ws_size;
    const float*     pos   = (const float*)d_in[0];      // [N,3] f32
    const long long* batch = (const long long*)d_in[1];  // [N] i64 (sorted)
    float*           out   = (float*)d_out;
    int N = in_sizes[0] / 3;
    int tiles = (N + 15) / 16;
    radius_graph_topk_kernel<<<tiles, 32, 0, stream>>>(pos, batch, out, N);
}